// WhiskerEncoder_53566832116137
// MI455X (gfx1250) — compile-verified
//
#include <hip/hip_runtime.h>
#include <hip/hip_bf16.h>

// MI455X (gfx1250) implementation.
// Roofline: after collapsing mean_t into the second linear layer, the problem is
// memory-bound on reading whisker_input (31.5 MB -> ~1.4us at 23.3 TB/s).
// All matmuls use V_WMMA_F32_16X16X4_F32 (K=4 matches IN=4 exactly; exact f32).

typedef float v2f __attribute__((ext_vector_type(2)));
typedef float v8f __attribute__((ext_vector_type(8)));

// Valid whisker indices (n in 0..59 with n%30 not in {4,9,16,23,29}); 50 total.
__constant__ int kValidN[50] = {
     0, 1, 2, 3, 5, 6, 7, 8,10,11,12,13,14,15,17,18,19,20,21,22,24,25,26,27,28,
    30,31,32,33,35,36,37,38,40,41,42,43,44,45,47,48,49,50,51,52,54,55,56,57,58
};

// ---------------------------------------------------------------------------
// Kernel 1: hs[b,j,32] = sum_t relu(x[b,n_j,t,:] @ W1[n_j] + b1[n_j])
// One block per (b, valid-n). 4 waves x 128 tokens each; 16-token WMMA tiles.
// ---------------------------------------------------------------------------
__global__ __launch_bounds__(128) void k1_hsum(
    const float* __restrict__ x,   // [64,60,512,4]
    const float* __restrict__ W1,  // [60,4,32]
    const float* __restrict__ b1,  // [60,32]
    float* __restrict__ hs)        // [64,50,32]
{
    const int bi   = blockIdx.x;          // 0..3199
    const int b    = bi / 50;
    const int jn   = bi - b * 50;
    const int n    = kValidN[jn];
    const int tid  = threadIdx.x;
    const int wid  = tid >> 5;
    const int lane = tid & 31;
    const int lh   = lane >> 4;           // lane half: K offset selector
    const int lm   = lane & 15;
    const int k0   = lh << 1;             // 0 or 2

    const float* xb  = x  + (((size_t)b * 60 + n) * 512) * 4;
    const float* W1n = W1 + (size_t)n * 4 * 32;
    const float* b1n = b1 + (size_t)n * 32;

    // B operand (4x16 f32): lane half selects K pair, VGPR j = K=k0+j, col = lm.
    v2f B0, B1;
    B0.x = W1n[(k0 + 0) * 32 + lm];
    B0.y = W1n[(k0 + 1) * 32 + lm];
    B1.x = W1n[(k0 + 0) * 32 + 16 + lm];
    B1.y = W1n[(k0 + 1) * 32 + 16 + lm];

    // Bias as C matrix: constant along M, so all 8 C VGPRs = b1[col].
    const float bias0 = b1n[lm];
    const float bias1 = b1n[16 + lm];
    v8f C0 = {}, C1 = {};
    #pragma unroll
    for (int r = 0; r < 8; ++r) { C0[r] = bias0; C1[r] = bias1; }

    v8f acc0 = {}, acc1 = {};
    const int tbase = wid * 128;          // 128 tokens per wave

    #pragma unroll
    for (int c = 0; c < 8; ++c) {
        const int tok = tbase + c * 16 + lm;      // A: M = lm for both halves
        const float* xr = xb + (size_t)tok * 4 + k0;
        v2f A; A.x = xr[0]; A.y = xr[1];
        v8f d0 = __builtin_amdgcn_wmma_f32_16x16x4_f32(false, A, false, B0,
                                                       (short)0, C0, false, false);
        v8f d1 = __builtin_amdgcn_wmma_f32_16x16x4_f32(false, A, false, B1,
                                                       (short)0, C1, false, false);
        #pragma unroll
        for (int r = 0; r < 8; ++r) {
            acc0[r] += fmaxf(d0[r], 0.0f);
            acc1[r] += fmaxf(d1[r], 0.0f);
        }
    }

    // Reduce over M: sum 8 VGPRs (M=r / r+8 per half), then add the other half.
    float s0 = 0.0f, s1 = 0.0f;
    #pragma unroll
    for (int r = 0; r < 8; ++r) { s0 += acc0[r]; s1 += acc1[r]; }
    s0 += __shfl_xor(s0, 16, 32);
    s1 += __shfl_xor(s1, 16, 32);

    __shared__ float red[4][32];
    if (lane < 16) { red[wid][lm] = s0; red[wid][16 + lm] = s1; }
    __syncthreads();
    if (tid < 32) {
        const float v = red[0][tid] + red[1][tid] + red[2][tid] + red[3][tid];
        hs[((size_t)b * 50 + jn) * 32 + tid] = v;
    }
}

// ---------------------------------------------------------------------------
// Kernel 2: pooled[b,d] = (1/35) * ( (1/512) * sum_j hs[b,j,:]@W2[n_j,:,d]
//                                     + sum_j b2[n_j,d] )
// Grouped GEMM M=64, N=128, K=50*32 via chained K=4 WMMAs. 1 tile per wave.
// ---------------------------------------------------------------------------
__global__ __launch_bounds__(128) void k2_pooled(
    const float* __restrict__ hs,     // [64,50,32]
    const float* __restrict__ W2,     // [60,32,128]
    const float* __restrict__ b2,     // [60,128]
    float* __restrict__ pooled)       // [64,128]
{
    const int tid  = threadIdx.x;
    const int wid  = tid >> 5;
    const int lane = tid & 31;
    const int lh   = lane >> 4;
    const int lm   = lane & 15;
    const int k0   = lh << 1;
    const int tile = blockIdx.x * 4 + wid;  // 0..31
    const int mt   = tile >> 3;             // 4 M tiles (b rows)
    const int nt   = tile & 7;              // 8 N tiles (embed dims)
    const int d    = nt * 16 + lm;          // B/C/D column
    const int m    = mt * 16 + lm;          // A row (all lanes)

    v8f C = {};
    for (int j = 0; j < 50; ++j) {
        const int n = kValidN[j];
        const float* hsr = hs + ((size_t)m * 50 + j) * 32;
        const float* w2n = W2 + (size_t)n * 32 * 128;
        #pragma unroll
        for (int kt = 0; kt < 8; ++kt) {
            const int k = kt * 4 + k0;
            v2f A; A.x = hsr[k];             A.y = hsr[k + 1];
            v2f B; B.x = w2n[k * 128 + d];   B.y = w2n[(k + 1) * 128 + d];
            C = __builtin_amdgcn_wmma_f32_16x16x4_f32(false, A, false, B,
                                                      (short)0, C, false, false);
        }
    }

    float bsum = 0.0f;
    for (int j = 0; j < 50; ++j) bsum += b2[(size_t)kValidN[j] * 128 + d];

    const float invT  = 1.0f / 512.0f;
    const float inv35 = 1.0f / 35.0f;
    #pragma unroll
    for (int r = 0; r < 8; ++r) {
        const int row = mt * 16 + r + 8 * lh;        // D layout: M = r / r+8
        pooled[(size_t)row * 128 + d] = (C[r] * invT + bsum) * inv35;
    }
}

// ---------------------------------------------------------------------------
// Kernel 3: out = pooled @ Wout^T + bout.  M=64, N=128, K=128.
// ---------------------------------------------------------------------------
__global__ __launch_bounds__(128) void k3_out(
    const float* __restrict__ pooled, // [64,128]
    const float* __restrict__ Wout,   // [128,128]
    const float* __restrict__ bout,   // [128]
    float* __restrict__ out)          // [64,128]
{
    const int tid  = threadIdx.x;
    const int wid  = tid >> 5;
    const int lane = tid & 31;
    const int lh   = lane >> 4;
    const int lm   = lane & 15;
    const int k0   = lh << 1;
    const int tile = blockIdx.x * 4 + wid;
    const int mt   = tile >> 3;
    const int nt   = tile & 7;
    const int d    = nt * 16 + lm;
    const int m    = mt * 16 + lm;

    const float bv = bout[d];
    v8f C = {};
    #pragma unroll
    for (int r = 0; r < 8; ++r) C[r] = bv;

    #pragma unroll 4
    for (int kt = 0; kt < 32; ++kt) {
        const int k = kt * 4 + k0;
        // A[m][k..k+1] and B[k..k+1][d] = Wout[d][k..k+1] (contiguous pairs)
        v2f A = *(const v2f*)(pooled + (size_t)m * 128 + k);
        v2f B = *(const v2f*)(Wout   + (size_t)d * 128 + k);
        C = __builtin_amdgcn_wmma_f32_16x16x4_f32(false, A, false, B,
                                                  (short)0, C, false, false);
    }

    #pragma unroll
    for (int r = 0; r < 8; ++r) {
        const int row = mt * 16 + r + 8 * lh;
        out[(size_t)row * 128 + d] = C[r];
    }
}

extern "C" void kernel_launch(void* const* d_in, const int* in_sizes, int n_in,
                              void* d_out, int out_size, void* d_ws, size_t ws_size,
                              hipStream_t stream) {
    const float* x    = (const float*)d_in[0]; // [64,60,512,4]
    const float* W1   = (const float*)d_in[1]; // [60,4,32]
    const float* b1   = (const float*)d_in[2]; // [60,32]
    const float* W2   = (const float*)d_in[3]; // [60,32,128]
    const float* b2   = (const float*)d_in[4]; // [60,128]
    const float* Wout = (const float*)d_in[5]; // [128,128]
    const float* bout = (const float*)d_in[6]; // [128]
    float* out = (float*)d_out;                // [64,128]

    float* hs     = (float*)d_ws;                          // 64*50*32 f32 = 400 KB
    float* pooled = (float*)((char*)d_ws + 64 * 50 * 32 * sizeof(float));

    k1_hsum  <<<64 * 50, 128, 0, stream>>>(x, W1, b1, hs);
    k2_pooled<<<8,       128, 0, stream>>>(hs, W2, b2, pooled);
    k3_out   <<<8,       128, 0, stream>>>(pooled, Wout, bout, out);
}